// MFAEBlock_69956427317493
// MI455X (gfx1250) — compile-verified
//
#include <hip/hip_runtime.h>
#include <hip/hip_bf16.h>
#include <math.h>

// ---------------------------------------------------------------------------
// MFAE block for MI455X (gfx1250). fp32 end-to-end.
//  - 1x1 convs   -> V_WMMA_F32_16X16X4_F32 (fp32 WMMA, exact vs reference)
//  - reparam convs merged on-device into single 9x9 / 19x19 depthwise kernels
//  - depthwise convs: async global->LDS tile staging + register-blocked FMA
// ---------------------------------------------------------------------------

#define EPSF 1e-5f
#define NPIX 65536            // 256*256
#define NB   4

typedef __attribute__((ext_vector_type(2))) float v2f;
typedef __attribute__((ext_vector_type(8))) float v8f;
typedef int v4i __attribute__((__vector_size__(16)));   // matches builtin param

// ---- gfx1250 async global->LDS path (ASYNCcnt), guarded by __has_builtin ---
#if defined(__has_builtin)
#  if __has_builtin(__builtin_amdgcn_global_load_async_to_lds_b32) && \
      __has_builtin(__builtin_amdgcn_global_load_async_to_lds_b128) && \
      __has_builtin(__builtin_amdgcn_s_wait_asynccnt)
#    define USE_ASYNC_LDS 1
#  endif
#endif
#ifndef USE_ASYNC_LDS
#  define USE_ASYNC_LDS 0
#endif

#if USE_ASYNC_LDS
__device__ __forceinline__ void async_b32(const float* g, float* l) {
  __builtin_amdgcn_global_load_async_to_lds_b32(
      (__attribute__((address_space(1))) int*)g,
      (__attribute__((address_space(3))) int*)l, 0, 0);
}
__device__ __forceinline__ void async_b128(const float* g, float* l) {
  __builtin_amdgcn_global_load_async_to_lds_b128(
      (__attribute__((address_space(1))) v4i*)g,
      (__attribute__((address_space(3))) v4i*)l, 0, 0);
}
__device__ __forceinline__ void async_wait0() {
  __builtin_amdgcn_s_wait_asynccnt(0);
}
#endif

__device__ __forceinline__ v8f wmma_f32_k4(v2f a, v2f b, v8f c) {
  // D = A(16x4) * B(4x16) + C(16x16), fp32
  return __builtin_amdgcn_wmma_f32_16x16x4_f32(false, a, false, b, (short)0, c,
                                               false, false);
}

// ---------------------------------------------------------------------------
// Kernel 0: fold BN into branches and merge reparam branches into one
// effective depthwise kernel (exact: all branches are "same"-padded).
//   rc2: 9x9 (pad4) += 3x3 @ offset 3, 5x5 @ offset 2
//   rc3: 19x19 (pad9) += 5x5 @ offset 7, 9x9 dil2 @ offsets 1+2i
// ---------------------------------------------------------------------------
__global__ void k_merge(
    const float* lk2w, const float* g2, const float* b2, const float* m2, const float* v2,
    const float* w2a, const float* w2b,
    const float* g2a, const float* b2a, const float* m2a, const float* v2a,
    const float* g2b, const float* b2b, const float* m2b, const float* v2b,
    const float* lk3w, const float* g3, const float* b3, const float* m3, const float* v3,
    const float* w3a, const float* w3b,
    const float* g3a, const float* b3a, const float* m3a, const float* v3a,
    const float* g3b, const float* b3b, const float* m3b, const float* v3b,
    float* mw2, float* mb2, float* mw3, float* mb3) {
  int c = threadIdx.x;
  if (c < 96) {
    float s0 = g2[c] * rsqrtf(v2[c] + EPSF);
    float sa = g2a[c] * rsqrtf(v2a[c] + EPSF);
    float sb = g2b[c] * rsqrtf(v2b[c] + EPSF);
    for (int i = 0; i < 81; ++i) mw2[c * 81 + i] = lk2w[c * 81 + i] * s0;
    for (int i = 0; i < 3; ++i)
      for (int j = 0; j < 3; ++j)
        mw2[c * 81 + (3 + i) * 9 + (3 + j)] += w2a[c * 9 + i * 3 + j] * sa;
    for (int i = 0; i < 5; ++i)
      for (int j = 0; j < 5; ++j)
        mw2[c * 81 + (2 + i) * 9 + (2 + j)] += w2b[c * 25 + i * 5 + j] * sb;
    mb2[c] = (b2[c] - m2[c] * s0) + (b2a[c] - m2a[c] * sa) + (b2b[c] - m2b[c] * sb);
  }
  if (c < 64) {
    float s0 = g3[c] * rsqrtf(v3[c] + EPSF);
    float sa = g3a[c] * rsqrtf(v3a[c] + EPSF);
    float sb = g3b[c] * rsqrtf(v3b[c] + EPSF);
    for (int i = 0; i < 361; ++i) mw3[c * 361 + i] = lk3w[c * 361 + i] * s0;
    for (int i = 0; i < 5; ++i)
      for (int j = 0; j < 5; ++j)
        mw3[c * 361 + (7 + i) * 19 + (7 + j)] += w3a[c * 25 + i * 5 + j] * sa;
    for (int i = 0; i < 9; ++i)      // dilation-2 taps
      for (int j = 0; j < 9; ++j)
        mw3[c * 361 + (1 + 2 * i) * 19 + (1 + 2 * j)] += w3b[c * 81 + i * 9 + j] * sb;
    mb3[c] = (b3[c] - m3[c] * s0) + (b3a[c] - m3a[c] * sa) + (b3b[c] - m3b[c] * sb);
  }
}

// ---------------------------------------------------------------------------
// Kernel 1: proj_in GEMM via WMMA fp32.  y[b,o,p] = sum_c W[o,c] x[b,c,p] + bias
// x tile (64ch x 16px) staged once per block into LDS (async b128), then
// 8 waves each produce one 16-out-channel tile; stores coalesced.
// ---------------------------------------------------------------------------
__global__ void k_proj_in(const float* __restrict__ x, const float* __restrict__ W,
                          const float* __restrict__ bias, float* __restrict__ y) {
  __shared__ float xt[64 * 16];        // [ch][px]
  const int tid = threadIdx.x;         // 256
  const int lane = tid & 31;
  const int wave = tid >> 5;           // out-channel tile 0..7
  const int pb = blockIdx.x * 16;      // pixel base within batch
  const int b = blockIdx.y;
  const float* xb = x + (size_t)b * 64 * NPIX;
#if USE_ASYNC_LDS
  {
    const int ch = tid >> 2, g = (tid & 3) * 4;   // 16B per thread
    async_b128(xb + (size_t)ch * NPIX + pb + g, &xt[ch * 16 + g]);
    async_wait0();
  }
#else
  for (int i = tid; i < 1024; i += 256) {
    const int ch = i >> 4, px = i & 15;
    xt[ch * 16 + px] = xb[(size_t)ch * NPIX + pb + px];
  }
#endif
  __syncthreads();

  const int n = lane & 15;
  const int hi = lane >> 4;
  v8f acc = {};
#pragma unroll
  for (int kb = 0; kb < 64; kb += 4) {
    const int k0 = kb + 2 * hi;
    v2f a, bv;
    a.x = W[(wave * 16 + n) * 64 + k0];
    a.y = W[(wave * 16 + n) * 64 + k0 + 1];
    bv.x = xt[k0 * 16 + n];
    bv.y = xt[(k0 + 1) * 16 + n];
    acc = wmma_f32_k4(a, bv, acc);
  }
  float* yb = y + (size_t)b * 128 * NPIX;
#pragma unroll
  for (int r = 0; r < 8; ++r) {
    const int ch = wave * 16 + r + 8 * hi;
    yb[(size_t)ch * NPIX + pb + n] = acc[r] + bias[ch];
  }
}

// ---------------------------------------------------------------------------
// Kernel 2: depthwise 3x3 + bias + residual on y channels 16..127 -> dw_abc
// ---------------------------------------------------------------------------
__global__ void k_dw3x3(const float* __restrict__ y, const float* __restrict__ w,
                        const float* __restrict__ bias, float* __restrict__ out) {
  const int idx = blockIdx.x * blockDim.x + threadIdx.x;  // 4*112*65536
  const int p = idx & (NPIX - 1);
  const int cz = idx >> 16;                 // b*112 + c
  const int c = cz % 112, b = cz / 112;
  const int h = p >> 8, wq = p & 255;
  const float* src = y + (size_t)(b * 128 + 16 + c) * NPIX;
  float acc = bias[c] + src[p];
#pragma unroll
  for (int i = 0; i < 3; ++i)
#pragma unroll
    for (int j = 0; j < 3; ++j) {
      const int hh = h + i - 1, ww = wq + j - 1;
      if (hh >= 0 && hh < 256 && ww >= 0 && ww < 256)
        acc += src[hh * 256 + ww] * w[c * 9 + i * 3 + j];
    }
  out[(size_t)cz * NPIX + p] = acc;
}

// ---------------------------------------------------------------------------
// Kernel 3/4: merged depthwise KxK (K=9 or 19).
// 64x16 output tile per block; each thread computes a 1x4 strip with a sliding
// 3-register window (~(K+3) LDS reads per row for 4 outputs -> 4:1 VALU:ds).
// Interior tiles staged with gfx1250 async global->LDS; border tiles use the
// zero-padded manual fill.
// ---------------------------------------------------------------------------
template <int K, int CIN_OFF, int CIN_TOT, int COUT>
__global__ void k_dwK(const float* __restrict__ in, const float* __restrict__ w,
                      const float* __restrict__ bias, float* __restrict__ out) {
  constexpr int R = K / 2;
  constexpr int TW = 64 + 2 * R;
  constexpr int TH = 16 + 2 * R;
  __shared__ float tile[TW * TH];
  const int c = blockIdx.z % COUT, b = blockIdx.z / COUT;
  const int oy0 = blockIdx.y * 16, ox0 = blockIdx.x * 64;
  const float* src = in + (size_t)(b * CIN_TOT + CIN_OFF + c) * NPIX;
  const int tid = threadIdx.x;  // 256

  const bool interior = (oy0 >= R) && (oy0 + 16 + R <= 256) &&
                        (ox0 >= R) && (ox0 + 64 + R <= 256);
#if USE_ASYNC_LDS
  if (interior) {
    for (int i = tid; i < TW * TH; i += 256) {
      const int ty = i / TW, tx = i % TW;
      async_b32(src + (oy0 - R + ty) * 256 + (ox0 - R + tx), &tile[i]);
    }
    async_wait0();
  } else
#endif
  {
    (void)interior;
    for (int i = tid; i < TW * TH; i += 256) {
      const int ty = i / TW, tx = i % TW;
      const int gy = oy0 - R + ty, gx = ox0 - R + tx;
      tile[i] = (gy >= 0 && gy < 256 && gx >= 0 && gx < 256) ? src[gy * 256 + gx] : 0.0f;
    }
  }
  __syncthreads();

  const int ly = tid >> 4, lx = tid & 15;  // 16x16 threads, 1x4 outputs each
  const float* wc = w + c * K * K;
  const float bs = bias[c];
  float a0 = bs, a1 = bs, a2 = bs, a3 = bs;
  for (int i = 0; i < K; ++i) {
    const float* trow = &tile[(ly + i) * TW + lx * 4];
    float w0 = trow[0], w1 = trow[1], w2 = trow[2];
#pragma unroll
    for (int j = 0; j < K; ++j) {
      const float nv = trow[j + 3];
      const float wv = wc[i * K + j];
      a0 = fmaf(w0, wv, a0);
      a1 = fmaf(w1, wv, a1);
      a2 = fmaf(w2, wv, a2);
      a3 = fmaf(nv, wv, a3);
      w0 = w1; w1 = w2; w2 = nv;
    }
  }
  float4 r = make_float4(a0, a1, a2, a3);
  *(float4*)&out[(size_t)(b * COUT + c) * NPIX + (oy0 + ly) * 256 + ox0 + lx * 4] = r;
}

// ---------------------------------------------------------------------------
// Kernel 5: attention maps -> 3-way softmax gate (4,3,256,256)
// attn = [dw_abc[0:16], dw_cde[0:32], dw_fgh[0:64]]  (112 channels)
// ---------------------------------------------------------------------------
__global__ void k_amp(const float* __restrict__ dwabc, const float* __restrict__ dwcde,
                      const float* __restrict__ dwfgh,
                      const float* w11, const float* w12, const float* w21,
                      const float* w22, const float* w31, const float* w32,
                      float* __restrict__ soft) {
  const int idx = blockIdx.x * blockDim.x + threadIdx.x;  // 4*65536
  const int p = idx & (NPIX - 1), b = idx >> 16;
  const float* p1 = dwabc + (size_t)b * 112 * NPIX + p;
  const float* p2 = dwcde + (size_t)b * 96 * NPIX + p;
  const float* p3 = dwfgh + (size_t)b * 64 * NPIX + p;
  float mx = -INFINITY, sm = 0.f;
  for (int c = 0; c < 16; ++c) { float v = p1[(size_t)c * NPIX]; mx = fmaxf(mx, v); sm += v; }
  for (int c = 0; c < 32; ++c) { float v = p2[(size_t)c * NPIX]; mx = fmaxf(mx, v); sm += v; }
  for (int c = 0; c < 64; ++c) { float v = p3[(size_t)c * NPIX]; mx = fmaxf(mx, v); sm += v; }
  const float xa = sm * (1.0f / 112.0f);
  const float W1[3] = {*w11, *w21, *w31};
  const float W2[3] = {*w12, *w22, *w32};
  float s[3], smax = -INFINITY;
#pragma unroll
  for (int k = 0; k < 3; ++k) {
    const float f = W2[k] * fmaxf(W1[k] * xa, 0.f) + W2[k] * fmaxf(W1[k] * mx, 0.f);
    s[k] = 1.f / (1.f + __expf(-f));
    smax = fmaxf(smax, s[k]);
  }
  float den = 0.f;
#pragma unroll
  for (int k = 0; k < 3; ++k) { s[k] = __expf(s[k] - smax); den += s[k]; }
  const float inv = 1.f / den;
#pragma unroll
  for (int k = 0; k < 3; ++k)
    soft[(size_t)(b * 3 + k) * NPIX + p] = s[k] * inv;
}

// ---------------------------------------------------------------------------
// Kernel 6: fused gated 1x1-conv chain with WMMA.
// One wave owns a 16-pixel tile and chains 16->32->64->64 through LDS.
// ---------------------------------------------------------------------------
__global__ void k_final(const float* __restrict__ y, const float* __restrict__ dwabc,
                        const float* __restrict__ dwcde, const float* __restrict__ dwfgh,
                        const float* __restrict__ soft,
                        const float* __restrict__ pw0w, const float* __restrict__ pw0b,
                        const float* __restrict__ pw1w, const float* __restrict__ pw1b,
                        const float* __restrict__ pow_, const float* __restrict__ pob,
                        float* __restrict__ out) {
  __shared__ float g1[4][16 * 16];
  __shared__ float g2[4][32 * 16];
  __shared__ float g3[4][64 * 16];
  const int lane = threadIdx.x & 31, wave = threadIdx.x >> 5;
  const int gtile = blockIdx.x * 4 + wave;    // 16384 tiles total
  const int b = gtile >> 12;                  // /4096
  const int pb = (gtile & 4095) * 16;
  const int n = lane & 15, hi = lane >> 4;
  const float* pwa = y + (size_t)b * 128 * NPIX;       // channels 0..15
  const float* a1 = dwabc + (size_t)b * 112 * NPIX;    // channels 0..15
  const float* a2 = dwcde + (size_t)b * 96 * NPIX;     // channels 0..31
  const float* a3 = dwfgh + (size_t)b * 64 * NPIX;     // 64 channels
  const float* s0 = soft + (size_t)(b * 3 + 0) * NPIX + pb;
  const float* s1 = soft + (size_t)(b * 3 + 1) * NPIX + pb;
  const float* s2 = soft + (size_t)(b * 3 + 2) * NPIX + pb;

  // stage 0: g1 = pwa * attn1 * soft0   (16 ch x 16 px)
  for (int i = lane; i < 256; i += 32) {
    const int ch = i >> 4, px = i & 15;
    g1[wave][i] = pwa[(size_t)ch * NPIX + pb + px] *
                  a1[(size_t)ch * NPIX + pb + px] * s0[px];
  }
  __syncthreads();

  // GEMM1: (32x16) = pw0(32x16) @ g1, then gate with attn2*soft1
  for (int mt = 0; mt < 2; ++mt) {
    v8f acc = {};
#pragma unroll
    for (int kb = 0; kb < 16; kb += 4) {
      const int k0 = kb + 2 * hi;
      v2f a, bv;
      a.x = pw0w[(mt * 16 + n) * 16 + k0];
      a.y = pw0w[(mt * 16 + n) * 16 + k0 + 1];
      bv.x = g1[wave][k0 * 16 + n];
      bv.y = g1[wave][(k0 + 1) * 16 + n];
      acc = wmma_f32_k4(a, bv, acc);
    }
#pragma unroll
    for (int r = 0; r < 8; ++r) {
      const int ch = mt * 16 + r + 8 * hi;
      g2[wave][ch * 16 + n] =
          (acc[r] + pw0b[ch]) * a2[(size_t)ch * NPIX + pb + n] * s1[n];
    }
  }
  __syncthreads();

  // GEMM2: (64x16) = pw1(64x32) @ g2, gate with attn3*soft2
  for (int mt = 0; mt < 4; ++mt) {
    v8f acc = {};
#pragma unroll
    for (int kb = 0; kb < 32; kb += 4) {
      const int k0 = kb + 2 * hi;
      v2f a, bv;
      a.x = pw1w[(mt * 16 + n) * 32 + k0];
      a.y = pw1w[(mt * 16 + n) * 32 + k0 + 1];
      bv.x = g2[wave][k0 * 16 + n];
      bv.y = g2[wave][(k0 + 1) * 16 + n];
      acc = wmma_f32_k4(a, bv, acc);
    }
#pragma unroll
    for (int r = 0; r < 8; ++r) {
      const int ch = mt * 16 + r + 8 * hi;
      g3[wave][ch * 16 + n] =
          (acc[r] + pw1b[ch]) * a3[(size_t)ch * NPIX + pb + n] * s2[n];
    }
  }
  __syncthreads();

  // GEMM3: out = proj_out(64x64) @ g3 + bias  (coalesced stores per half-wave)
  float* ob = out + (size_t)b * 64 * NPIX;
  for (int mt = 0; mt < 4; ++mt) {
    v8f acc = {};
#pragma unroll
    for (int kb = 0; kb < 64; kb += 4) {
      const int k0 = kb + 2 * hi;
      v2f a, bv;
      a.x = pow_[(mt * 16 + n) * 64 + k0];
      a.y = pow_[(mt * 16 + n) * 64 + k0 + 1];
      bv.x = g3[wave][k0 * 16 + n];
      bv.y = g3[wave][(k0 + 1) * 16 + n];
      acc = wmma_f32_k4(a, bv, acc);
    }
#pragma unroll
    for (int r = 0; r < 8; ++r) {
      const int ch = mt * 16 + r + 8 * hi;
      ob[(size_t)ch * NPIX + pb + n] = acc[r] + pob[ch];
    }
  }
}

// ---------------------------------------------------------------------------
// d_in order (setup_inputs dict order, nested dicts/lists flattened):
//  0 x | 1 proj_in_w | 2 proj_in_b | 3 dw1_w | 4 dw1_b
//  5 rc2.lk_w | 6..9 rc2.lk_bn{g,b,m,v} | 10 rc2.br_w[0] | 11 rc2.br_w[1]
//  12..15 rc2.br_bn[0] | 16..19 rc2.br_bn[1]
//  20 rc3.lk_w | 21..24 rc3.lk_bn | 25 rc3.br_w[0] | 26 rc3.br_w[1]
//  27..30 rc3.br_bn[0] | 31..34 rc3.br_bn[1]
//  35 amp1.w1 | 36 amp1.w2 | 37 amp2.w1 | 38 amp2.w2 | 39 amp3.w1 | 40 amp3.w2
//  41 pw0_w | 42 pw0_b | 43 pw1_w | 44 pw1_b | 45 proj_out_w | 46 proj_out_b
// ---------------------------------------------------------------------------
extern "C" void kernel_launch(void* const* d_in, const int* in_sizes, int n_in,
                              void* d_out, int out_size, void* d_ws, size_t ws_size,
                              hipStream_t stream) {
  const float* F[47];
  for (int i = 0; i < 47 && i < n_in; ++i) F[i] = (const float*)d_in[i];
  float* out = (float*)d_out;

  // workspace layout (floats) — ~423 MB total
  float* ws = (float*)d_ws;
  float* y = ws;                                   // 4*128*65536
  float* dwabc = y + (size_t)NB * 128 * NPIX;      // 4*112*65536
  float* dwcde = dwabc + (size_t)NB * 112 * NPIX;  // 4*96*65536
  float* dwfgh = dwcde + (size_t)NB * 96 * NPIX;   // 4*64*65536
  float* soft = dwfgh + (size_t)NB * 64 * NPIX;    // 4*3*65536
  float* mw2 = soft + (size_t)NB * 3 * NPIX;       // 96*81
  float* mb2 = mw2 + 96 * 81;                      // 96
  float* mw3 = mb2 + 96;                           // 64*361
  float* mb3 = mw3 + 64 * 361;                     // 64

  // 0) fold BN + merge reparam branches
  k_merge<<<1, 96, 0, stream>>>(
      F[5], F[6], F[7], F[8], F[9], F[10], F[11],
      F[12], F[13], F[14], F[15], F[16], F[17], F[18], F[19],
      F[20], F[21], F[22], F[23], F[24], F[25], F[26],
      F[27], F[28], F[29], F[30], F[31], F[32], F[33], F[34],
      mw2, mb2, mw3, mb3);

  // 1) proj_in GEMM (WMMA fp32): 16-pixel tiles x 8 out-channel tiles
  k_proj_in<<<dim3(NPIX / 16, NB), 256, 0, stream>>>(F[0], F[1], F[2], y);

  // 2) depthwise 3x3 + bias + residual
  k_dw3x3<<<(NB * 112 * NPIX) / 256, 256, 0, stream>>>(y, F[3], F[4], dwabc);

  // 3) merged rc2 9x9 on dw_abc[16:112] -> dw_cde  (64x16 tiles)
  k_dwK<9, 16, 112, 96><<<dim3(4, 16, NB * 96), 256, 0, stream>>>(dwabc, mw2, mb2, dwcde);

  // 4) merged rc3 19x19 on dw_cde[32:96] -> dw_fgh (64x16 tiles)
  k_dwK<19, 32, 96, 64><<<dim3(4, 16, NB * 64), 256, 0, stream>>>(dwcde, mw3, mb3, dwfgh);

  // 5) attention gates
  k_amp<<<(NB * NPIX) / 256, 256, 0, stream>>>(dwabc, dwcde, dwfgh,
                                               F[35], F[36], F[37], F[38], F[39], F[40],
                                               soft);

  // 6) fused gated 1x1-conv chain (WMMA)
  k_final<<<(NB * NPIX / 16) / 4, 128, 0, stream>>>(y, dwabc, dwcde, dwfgh, soft,
                                                    F[41], F[42], F[43], F[44],
                                                    F[45], F[46], out);
}